// SIGformer_77378130805160
// MI455X (gfx1250) — compile-verified
//
#include <hip/hip_runtime.h>
#include <hip/hip_bf16.h>

typedef __attribute__((ext_vector_type(2))) float v2f;
typedef __attribute__((ext_vector_type(8))) float v8f;

#define HID 64
#define LN_EPS 1e-5f

// -------- helpers --------

__device__ __forceinline__ void atomicAddF(float* p, float v) {
  __hip_atomic_fetch_add(p, v, __ATOMIC_RELAXED, __HIP_MEMORY_SCOPE_AGENT);
}

// select c[i] with i in [0,8) using a cndmask chain (constant indices after unroll)
__device__ __forceinline__ float vsel8(v8f c, int i) {
  float r = c[0];
#pragma unroll
  for (int k = 1; k < 8; ++k) r = (i == k) ? c[k] : r;
  return r;
}

// -------- kernel 1: layernorm (one node per wave32) --------

__global__ void sig_ln_kernel(const float* __restrict__ x,
                              float* __restrict__ y, int n) {
  int wave = (int)((blockIdx.x * blockDim.x + threadIdx.x) >> 5);
  int lane = threadIdx.x & 31;
  if (wave >= n) return;
  const float* row = x + (size_t)wave * HID;
  float2 v = *(const float2*)(row + lane * 2);
  float s  = v.x + v.y;
  float s2 = v.x * v.x + v.y * v.y;
#pragma unroll
  for (int m = 16; m; m >>= 1) {
    s  += __shfl_xor(s,  m, 32);
    s2 += __shfl_xor(s2, m, 32);
  }
  float mu  = s * (1.0f / HID);
  float var = s2 * (1.0f / HID) - mu * mu;
  float inv = rsqrtf(var + LN_EPS);
  float2 o;
  o.x = (v.x - mu) * inv;
  o.y = (v.y - mu) * inv;
  *(float2*)(y + (size_t)wave * HID + lane * 2) = o;
}

// -------- kernel 2: edge scores via V_WMMA_F32_16X16X4_F32 --------
// One wave handles 16 edges. A = y[dst] rows (16x4 slab), B = y[src] cols
// (4x16 slab). Diagonal of the 16x16 accumulator = per-edge dot products.
// A 16x4 f32 layout: lane L holds row (L&15), K = k0 + (L<16 ? {0,1} : {2,3}).
// B 4x16 f32 layout: lane L holds col (L&15), same K split -> identical
// per-lane addressing, just dst vs src rows.
// C/D 16x16 f32: VGPR v of lane L holds M = v + (L<16?0:8), N = L&15.
// Diagonal m: (lane m, c[m]) for m<8 ; (lane m+16, c[m-8]) for m>=8.

__global__ void sig_score_kernel(const float* __restrict__ Y,
                                 const float* __restrict__ eigs,
                                 const float* __restrict__ lambda0,
                                 const float* __restrict__ path_emb,
                                 const int* __restrict__ dstA,
                                 const int* __restrict__ srcA,
                                 const int* __restrict__ ptype,
                                 float* __restrict__ E0, float* __restrict__ E1,
                                 float* __restrict__ D0, float* __restrict__ D1,
                                 int E) {
  int lane = threadIdx.x & 31;
  int wave = (int)((blockIdx.x * blockDim.x + threadIdx.x) >> 5);
  long base = (long)wave * 16;
  if (base >= E) return;  // wave-uniform exit: EXEC stays all-1s for WMMA

  int e  = (int)base + (lane & 15);
  int ec = (e < E) ? e : (E - 1);  // clamp reads; writes guarded below
  int d = dstA[ec];
  int s = srcA[ec];
  const float* yd = Y    + (size_t)d * HID;
  const float* ys = Y    + (size_t)s * HID;
  const float* ed = eigs + (size_t)d * HID;
  const float* es = eigs + (size_t)s * HID;
  int kb = (lane < 16) ? 0 : 2;

  v8f cy = {};
  v8f ce = {};
#pragma unroll
  for (int k0 = 0; k0 < HID; k0 += 4) {
    v2f a  = *(const v2f*)(yd + k0 + kb);
    v2f b  = *(const v2f*)(ys + k0 + kb);
    cy = __builtin_amdgcn_wmma_f32_16x16x4_f32(false, a, false, b,
                                               (short)0, cy, false, false);
    v2f ae = *(const v2f*)(ed + k0 + kb);
    v2f be = *(const v2f*)(es + k0 + kb);
    ce = __builtin_amdgcn_wmma_f32_16x16x4_f32(false, ae, false, be,
                                               (short)0, ce, false, false);
  }

  bool owner = (lane < 8) || (lane >= 24);
  int  ci    = (lane < 8) ? lane : (lane - 24);
  float dqk = vsel8(cy, ci);  // y[dst]·y[src] for this lane's edge
  float dei = vsel8(ce, ci);  // eigs[dst]·eigs[src]

  if (owner && e < E) {
    float s0  = dqk * 0.125f + __expf(lambda0[0]) * dei;  // 1/sqrt(64)=0.125
    float e0v = fminf(__expf(s0), 5.0f);                  // clip(exp,-5,5), exp>0
    float pe  = path_emb[ptype[e]];
    float e1v = fminf(__expf(pe), 5.0f);
    E0[e] = e0v;
    E1[e] = e1v;
    atomicAddF(&D0[d], e0v);
    atomicAddF(&D1[d], e1v);
  }
}

// -------- kernel 3: normalize edge weights (overwrites E0 with attn) --------

__global__ void sig_attn_kernel(float* __restrict__ E0,
                                const float* __restrict__ E1,
                                const float* __restrict__ D0,
                                const float* __restrict__ D1,
                                const int* __restrict__ dstA, int E) {
  int t = blockIdx.x * blockDim.x + threadIdx.x;
  if (t >= E) return;
  int d = dstA[t];
  float d0 = D0[d]; d0 = (d0 == 0.0f) ? 1.0f : d0;
  float d1 = D1[d]; d1 = (d1 == 0.0f) ? 1.0f : d1;
  E0[t] = 0.5f * (E0[t] / d0 + E1[t] / d1);
}

// -------- kernel 4: SPMM scatter  out[dst] += attn * y[src] --------
// thread = (edge, dim): consecutive threads cover one row -> coalesced
// gathers of y[src] and coalesced atomic scatters to out[dst].

__global__ void sig_spmm_kernel(const float* __restrict__ Y,
                                const float* __restrict__ attn,
                                const int* __restrict__ dstA,
                                const int* __restrict__ srcA,
                                float* __restrict__ out, long total) {
  long t = (long)blockIdx.x * blockDim.x + threadIdx.x;
  if (t >= total) return;
  int e   = (int)(t >> 6);
  int dim = (int)(t & 63);
  int d = dstA[e];
  int s = srcA[e];
  atomicAddF(out + (size_t)d * HID + dim, attn[e] * Y[(size_t)s * HID + dim]);
}

// -------- launch --------

extern "C" void kernel_launch(void* const* d_in, const int* in_sizes, int n_in,
                              void* d_out, int out_size, void* d_ws, size_t ws_size,
                              hipStream_t stream) {
  const float* x        = (const float*)d_in[0];
  const float* eigs     = (const float*)d_in[1];
  const float* lambda0  = (const float*)d_in[2];
  const float* path_emb = (const float*)d_in[3];
  const int*   indices  = (const int*)d_in[4];   // [2, E] (harness int ABI)
  const int*   ptype    = (const int*)d_in[5];   // [E]

  const int N = in_sizes[0] / HID;
  const int E = in_sizes[5];
  const int* dstA = indices;       // row 0
  const int* srcA = indices + E;   // row 1

  // workspace layout (floats): Y[N*64] | E0[E] | E1[E] | D0[N] | D1[N]
  float* Y  = (float*)d_ws;
  float* E0 = Y  + (size_t)N * HID;
  float* E1 = E0 + (size_t)E;
  float* D0 = E1 + (size_t)E;
  float* D1 = D0 + (size_t)N;

  // zero the atomic accumulators and output every call (graph-safe memsets)
  hipMemsetAsync(D0, 0, (size_t)N * 2 * sizeof(float), stream);  // D0 and D1
  hipMemsetAsync(d_out, 0, (size_t)out_size * sizeof(float), stream);

  // 1) layernorm: one node per wave, 8 waves per 256-thread block
  {
    int blocks = (N + 7) / 8;
    sig_ln_kernel<<<blocks, 256, 0, stream>>>(x, Y, N);
  }
  // 2) WMMA edge scores: 16 edges per wave, 8 waves per block
  {
    long waves  = ((long)E + 15) / 16;
    int  blocks = (int)((waves + 7) / 8);
    sig_score_kernel<<<blocks, 256, 0, stream>>>(Y, eigs, lambda0, path_emb,
                                                 dstA, srcA, ptype,
                                                 E0, E1, D0, D1, E);
  }
  // 3) normalize
  {
    int blocks = (E + 255) / 256;
    sig_attn_kernel<<<blocks, 256, 0, stream>>>(E0, E1, D0, D1, dstA, E);
  }
  // 4) SPMM scatter
  {
    long total  = (long)E * HID;
    long blocks = (total + 255) / 256;
    sig_spmm_kernel<<<(int)blocks, 256, 0, stream>>>(Y, E0, dstA, srcA,
                                                     (float*)d_out, total);
  }
}